// STPQuick_64003602645691
// MI455X (gfx1250) — compile-verified
//
#include <hip/hip_runtime.h>
#include <cstdint>
#include <math.h>

// STP forward as a first-order affine recurrence on r = imu/mu:
//   g[t] = exp(-(a + (x[t-1]+x[t])/2)),  c[t] = g[t]*x[t-1]/2 + x[t]/2
//   r[t] = g[t]*r[t-1] + c[t],  td[t] = (r>0) ? 1-r : 1
// Chunk-boundary clamp every 500 steps: r -> (r>0 ? r : 1)  (matches imu_s/mu_s carry).
// One WG per (b,u) chain. Per chunk: TDM (tensor_load_to_lds) stages the strided
// input slice into LDS (double-buffered, wave 0 issues, TENSORcnt-tracked), then a
// hierarchical wave32 shuffle scan composes the affine maps (3 barriers/chunk).

#define T_LEN   50000
#define U_CH    16
#define B_SZ    8
#define CHUNK   500
#define NCHUNK  100
#define BLK     512
#define NWAVES  (BLK / 32)

typedef unsigned int u32;
typedef __attribute__((ext_vector_type(4))) u32 u32x4;
typedef __attribute__((ext_vector_type(8))) u32 u32x8;

// TDM descriptor load: 2-D tile, 500 rows x 1 element (4B), row stride 16 elements,
// packed contiguously into LDS at lds_off. Issued once per wave (EXEC ignored).
__device__ __forceinline__ void tdm_load_chunk(const float* gsrc, u32 lds_off) {
  unsigned long long ga = (unsigned long long)(uintptr_t)gsrc;
  u32x4 g0;
  g0.x = 1u;                                            // count=1 (valid), no gather
  g0.y = lds_off;                                       // lds_addr (bytes)
  g0.z = (u32)ga;                                       // global_addr[31:0]
  g0.w = (u32)((ga >> 32) & 0x1FFFFFFull) | (2u << 30); // global_addr[56:32] | type=2
  u32x8 g1;
  g1.s0 = (2u << 16);        // workgroup_mask=0, data_size=2 (4 bytes), no pad/iter
  g1.s1 = (16u << 16);       // tensor_dim0 = 16 (U channels)
  g1.s2 = (50000u << 16);    // tensor_dim0[31:16]=0 | tensor_dim1[15:0] = 50000
  g1.s3 = (1u << 16);        // tensor_dim1[31:16]=0 | tile_dim0 = 1
  g1.s4 = 500u;              // tile_dim1 = 500, tile_dim2 = 0
  g1.s5 = 16u;               // tensor_dim0_stride = 16 (elements)
  g1.s6 = 0u;                // stride hi / tensor_dim1_stride (unused, 2D tile)
  g1.s7 = 0u;
  asm volatile("tensor_load_to_lds %0, %1" :: "s"(g0), "s"(g1) : "memory");
}

__device__ __forceinline__ void wait_tensor0() {
  __builtin_amdgcn_s_wait_tensorcnt(0);
  asm volatile("" ::: "memory");
}

__global__ __launch_bounds__(BLK, 1) void stp_scan_kernel(
    const float* __restrict__ inputs,   // (B, T, U)
    const float* __restrict__ u,        // (U,)
    const float* __restrict__ tau,      // (U,)
    float* __restrict__ out)            // (B, T, U)
{
  __shared__ float xs[2][CHUNK];        // TDM destination, double-buffered
  __shared__ float wg[NWAVES], wc[NWAVES];
  __shared__ float carry[3];            // {x_last, r_clamped, td_last}

  const int chain = blockIdx.x;         // 0..127
  const int b    = chain >> 4;
  const int uch  = chain & 15;
  const int i    = threadIdx.x;
  const int lane = i & 31;
  const int w    = i >> 5;

  // ui = |u|/FS*100 = |u| ; taui = max(|tau|, 2.001/FS)*FS ; a = 1/taui
  const float ui   = fabsf(u[uch]);
  const float taui = fmaxf(fabsf(tau[uch]), 2.001f / 100.0f) * 100.0f;
  const float a    = 1.0f / taui;

  const float* base  = inputs + ((size_t)b * T_LEN) * U_CH + uch;
  float*       obase = out    + ((size_t)b * T_LEN) * U_CH + uch;

  if (i == 0) { carry[0] = 0.0f; carry[1] = 0.0f; carry[2] = 1.0f; }

  if (w == 0) {                         // wave 0 owns the TDM pipeline
    tdm_load_chunk(base, (u32)(uintptr_t)&xs[0][0]);
    wait_tensor0();
  }
  __syncthreads();

  for (int j = 0; j < NCHUNK; ++j) {
    const int p = j & 1;

    // Prefetch next chunk via TDM (overlaps with the scan below).
    if (w == 0 && (j + 1) < NCHUNK)
      tdm_load_chunk(base + (size_t)(j + 1) * CHUNK * U_CH,
                     (u32)(uintptr_t)&xs[p ^ 1][0]);

    const float x0c  = carry[0];
    const float r0c  = carry[1];
    const float tdpc = carry[2];

    float g = 1.0f, c = 0.0f, tstim = 0.0f;
    if (i < CHUNK) {
      const float v = xs[p][i];
      tstim = (v != v) ? 0.0f : v;                 // NaN -> 0
      const float xi = ui * tstim;
      float xim1 = x0c;
      if (i > 0) {
        const float vp = xs[p][i - 1];
        xim1 = ui * ((vp != vp) ? 0.0f : vp);
      }
      g = __expf(-(a + 0.5f * (xim1 + xi)));       // v_exp_f32
      c = fmaf(g, 0.5f * xim1, 0.5f * xi);
    }

    // Intra-wave inclusive scan: (gR,cR) o (gL,cL) = (gR*gL, gR*cL + cR)
    #pragma unroll
    for (int off = 1; off < 32; off <<= 1) {
      const float gl = __shfl_up(g, off, 32);
      const float cl = __shfl_up(c, off, 32);
      if (lane >= off) { c = fmaf(g, cl, c); g = g * gl; }
    }
    if (lane == 31) { wg[w] = g; wc[w] = c; }
    __syncthreads();

    // Wave 0 scans the 16 wave composites.
    if (w == 0 && lane < NWAVES) {
      float gw = wg[lane], cw = wc[lane];
      #pragma unroll
      for (int off = 1; off < NWAVES; off <<= 1) {
        const float gl = __shfl_up(gw, off, 32);
        const float cl = __shfl_up(cw, off, 32);
        if (lane >= off) { cw = fmaf(gw, cl, cw); gw = gw * gl; }
      }
      wg[lane] = gw; wc[lane] = cw;
    }
    __syncthreads();

    // Apply wave prefix: total_i = local_i o prefix(w-1)
    const float gp = (w > 0) ? wg[w - 1] : 1.0f;
    const float cp = (w > 0) ? wc[w - 1] : 0.0f;
    const float gt = g * gp;
    const float ct = fmaf(g, cp, c);

    // Composite of element i-1 (for the td shift): shuffle, or wave prefix at lane 0.
    float gn = __shfl_up(gt, 1, 32);
    float cn = __shfl_up(ct, 1, 32);
    if (lane == 0) { gn = gp; cn = cp; }

    if (i < CHUNK) {
      const float r = fmaf(gt, r0c, ct);
      float tdsh;
      if (i == 0) {
        tdsh = tdpc;
      } else {
        const float rm1 = fmaf(gn, r0c, cn);
        tdsh = (rm1 > 0.0f) ? (1.0f - rm1) : 1.0f;
      }
      obase[((size_t)j * CHUNK + i) * U_CH] = tstim * tdsh;

      if (i == CHUNK - 1) {
        carry[0] = ui * tstim;                      // x carry
        carry[1] = (r > 0.0f) ? r : 1.0f;           // clamped ratio carry (imu_s/mu_s)
        carry[2] = (r > 0.0f) ? (1.0f - r) : 1.0f;  // td carry for the shift
      }
    }

    if (w == 0) wait_tensor0();   // next chunk resident in LDS
    __syncthreads();              // orders carry write, xs reads, TDM data for all waves
  }
}

extern "C" void kernel_launch(void* const* d_in, const int* in_sizes, int n_in,
                              void* d_out, int out_size, void* d_ws, size_t ws_size,
                              hipStream_t stream) {
  const float* inputs = (const float*)d_in[0];
  const float* u      = (const float*)d_in[1];
  const float* tau    = (const float*)d_in[2];
  float* out          = (float*)d_out;
  (void)in_sizes; (void)n_in; (void)out_size; (void)d_ws; (void)ws_size;

  stp_scan_kernel<<<dim3(B_SZ * U_CH), dim3(BLK), 0, stream>>>(inputs, u, tau, out);
}